// MyRNN_80221399154896
// MI455X (gfx1250) — compile-verified
//
#include <hip/hip_runtime.h>
#include <hip/hip_bf16.h>

// ---------------------------------------------------------------------------
// MI455X (gfx1250) implementation.
// Heavy work (logm, 256x batched expm, new_Ew) is done as f32 complex GEMMs
// built on V_WMMA_F32_16X16X4_F32 (the only f32 matrix op on CDNA5).
// Small O(u)/O(u^2) work stays in f64 VALU.
// GEMM: 64x64 block tile, K chunks of 16 staged in LDS with pair-interleaved
// layout (one ds_load_b64 per WMMA operand, conflict-free), register-staged
// double buffering of global loads (global_load_b128).
// ---------------------------------------------------------------------------

typedef __attribute__((ext_vector_type(2))) float v2f;
typedef __attribute__((ext_vector_type(8))) float v8f;

#define UDIM   256
#define IOFF   65536        // imag-plane offset (floats) inside a complex matrix
#define MATF   131072       // floats per complex matrix (2 planes)
#define TM     64
#define TN     64
#define TK     16
#define PP     160          // LDS pair-row pitch (dwords); 160 % 64 == 32

#define S2_SQ      8        // squarings, batched expm
#define S3_SQ      8        // squarings, new_uw expm
#define SLOG       5        // chained square roots in logm
#define NSIT       12       // Newton-Schulz iterations per sqrt level
#define MTAY       8        // Taylor (Horner) order for expm

// ---------------------------------------------------------------------------
// Kernel 1: f64 prologue. unit vector, 256-pt DFT + normalization, cell,
// update (output 0), log_diff.  One block of 256 threads.
// ---------------------------------------------------------------------------
__global__ __launch_bounds__(256)
void prep_kernel(const double* __restrict__ x,
                 const double* __restrict__ A_r, const double* __restrict__ A_i,
                 const double* __restrict__ cA_r, const double* __restrict__ cA_i,
                 double* __restrict__ unit_ws,    // [512] re/im interleaved
                 double* __restrict__ zscal,      // [512] ufft/2^S2 interleaved
                 double* __restrict__ logdif,     // [512]
                 float*  __restrict__ out_update) // [256]
{
    __shared__ double red[256];
    __shared__ double su_r[256], su_i[256];
    const int t = threadIdx.x;
    const double xk = x[t];

    // prod(x) via sum of logs
    red[t] = log(xk);
    __syncthreads();
    for (int s = 128; s > 0; s >>= 1) {
        if (t < s) red[t] += red[t + s];
        __syncthreads();
    }
    const double pinv = exp(-red[0] / 256.0);
    __syncthreads();

    const double dy = (t == 0) ? 0.0 : (xk - x[t - 1]);
    const double ur = cos(dy) * xk * pinv;
    const double ui = sin(dy) * xk * pinv;
    su_r[t] = ur; su_i[t] = ui;
    unit_ws[2 * t] = ur; unit_ws[2 * t + 1] = ui;
    __syncthreads();

    // naive 256-pt DFT (each thread one frequency)
    double fr = 0.0, fi = 0.0;
    const double w0 = -6.283185307179586476925286766559 * (double)t / 256.0;
    for (int n = 0; n < 256; ++n) {
        double c = cos(w0 * n), s = sin(w0 * n);
        fr += su_r[n] * c - su_i[n] * s;
        fi += su_r[n] * s + su_i[n] * c;
    }
    // sum over all frequencies
    red[t] = fr; __syncthreads();
    for (int s = 128; s > 0; s >>= 1) { if (t < s) red[t] += red[t + s]; __syncthreads(); }
    const double totr = red[0]; __syncthreads();
    red[t] = fi; __syncthreads();
    for (int s = 128; s > 0; s >>= 1) { if (t < s) red[t] += red[t + s]; __syncthreads(); }
    const double toti = red[0]; __syncthreads();

    const double den = totr * totr + toti * toti;
    const double ufr = (fr * totr + fi * toti) / den;
    const double ufi = (fi * totr - fr * toti) / den;
    const double inv2s2 = 1.0 / (double)(1 << S2_SQ);
    zscal[2 * t] = ufr * inv2s2;
    zscal[2 * t + 1] = ufi * inv2s2;

    // cell = conj(cellA) * (A * unit)
    const double ar = A_r[t], ai = A_i[t];
    const double tr = ar * ur - ai * ui;
    const double ti = ar * ui + ai * ur;
    const double ccr = cA_r[t], cci = cA_i[t];
    const double cr = ccr * tr + cci * ti;
    const double ci = ccr * ti - cci * tr;

    out_update[t] = (float)(xk + atan2(ci, cr));

    // log(cell) - log(unit)
    const double lr = 0.5 * (log(cr * cr + ci * ci) - log(ur * ur + ui * ui));
    const double li = atan2(ci, cr) - atan2(ui, ur);
    logdif[2 * t] = lr;
    logdif[2 * t + 1] = li;
}

// ---------------------------------------------------------------------------
// Kernel 2: new_uw = roll(uw,1,axis0).at[0] = log_diff.
// Writes output (float re/im interleaved) AND f32 planes scaled by 1/2^S3.
// ---------------------------------------------------------------------------
__global__ __launch_bounds__(256)
void build_uw_kernel(const double* __restrict__ uw_r, const double* __restrict__ uw_i,
                     const double* __restrict__ logdif,
                     float* __restrict__ out_newuw,   // interleaved floats [2*65536]
                     float* __restrict__ mXuw)        // planes, pre-scaled
{
    const int idx = blockIdx.x * 256 + threadIdx.x;
    const int row = idx >> 8, col = idx & 255;
    double re, im;
    if (row == 0) { re = logdif[2 * col]; im = logdif[2 * col + 1]; }
    else { re = uw_r[(row - 1) * 256 + col]; im = uw_i[(row - 1) * 256 + col]; }
    float2 w; w.x = (float)re; w.y = (float)im;
    *(float2*)&out_newuw[2 * idx] = w;
    const double sc = 1.0 / (double)(1 << S3_SQ);
    mXuw[idx]        = (float)(re * sc);
    mXuw[IOFF + idx] = (float)(im * sc);
}

// f64 matrix -> f32 planes
__global__ __launch_bounds__(256)
void convert_kernel(const double* __restrict__ mr, const double* __restrict__ mi,
                    float* __restrict__ dst)
{
    const int idx = blockIdx.x * 256 + threadIdx.x;
    dst[idx]        = (float)mr[idx];
    dst[IOFF + idx] = (float)mi[idx];
}

// trace-based normalization scalar for logm: c = |mean(diag(Ew))|
__global__ __launch_bounds__(256)
void trace_kernel(const double* __restrict__ mr, const double* __restrict__ mi,
                  double* __restrict__ scal)   // [0]=c [1]=1/c [2]=log c
{
    __shared__ double rr[256], ri[256];
    const int t = threadIdx.x;
    rr[t] = mr[t * 257]; ri[t] = mi[t * 257];
    __syncthreads();
    for (int s = 128; s > 0; s >>= 1) {
        if (t < s) { rr[t] += rr[t + s]; ri[t] += ri[t + s]; }
        __syncthreads();
    }
    if (t == 0) {
        double c = sqrt(rr[0] * rr[0] + ri[0] * ri[0]) / 256.0;
        c = fmax(c, 1e-30);
        scal[0] = c; scal[1] = 1.0 / c; scal[2] = log(c);
    }
}

// dst = src * (*scalePtr), over both planes
__global__ __launch_bounds__(256)
void scale_dev_kernel(float* __restrict__ dst, const float* __restrict__ src,
                      const double* __restrict__ scalePtr)
{
    const int idx = blockIdx.x * 256 + threadIdx.x;   // over 131072
    dst[idx] = src[idx] * (float)scalePtr[0];
}

// D = a*X + b*Y + g*I  (complex a,b; real g on diagonal). X or Y may be null.
__global__ __launch_bounds__(256)
void caxpby_kernel(float* __restrict__ D, const float* __restrict__ X,
                   const float* __restrict__ Y,
                   float ar, float ai, float br, float bi, float g)
{
    const int idx = blockIdx.x * 256 + threadIdx.x;   // over 65536
    const int row = idx >> 8, col = idx & 255;
    float dr = (row == col) ? g : 0.f, di = 0.f;
    if (X) {
        float xr = X[idx], xi = X[IOFF + idx];
        dr += ar * xr - ai * xi; di += ar * xi + ai * xr;
    }
    if (Y) {
        float yr = Y[idx], yi = Y[IOFF + idx];
        dr += br * yr - bi * yi; di += br * yi + bi * yr;
    }
    D[idx] = dr; D[IOFF + idx] = di;
}

// L += log(c) * I (real part of diagonal)
__global__ __launch_bounds__(256)
void add_logc_kernel(float* __restrict__ L, const double* __restrict__ scal)
{
    const int t = threadIdx.x;
    L[t * 257] += (float)scal[2];
}

// K = conj(L) - conj(L)^T
__global__ __launch_bounds__(256)
void build_K_kernel(const float* __restrict__ L, float* __restrict__ K)
{
    const int idx = blockIdx.x * 256 + threadIdx.x;
    const int r = idx >> 8, c = idx & 255;
    K[idx]        =  L[r * 256 + c]        - L[c * 256 + r];
    K[IOFF + idx] = -L[IOFF + r * 256 + c] + L[IOFF + c * 256 + r];
}

// P[b] = (z_b * invm) * K + I   (batched Horner seed)
__global__ __launch_bounds__(256)
void init_P_kernel(float* __restrict__ P, const float* __restrict__ K,
                   const double* __restrict__ zs, float invm)
{
    const int b = blockIdx.z;
    const int idx = blockIdx.x * 256 + threadIdx.x;
    const int row = idx >> 8, col = idx & 255;
    const float ar = (float)zs[2 * b] * invm, ai = (float)zs[2 * b + 1] * invm;
    const float kr = K[idx], ki = K[IOFF + idx];
    float* Pb = P + (long long)b * MATF;
    Pb[idx]        = ar * kr - ai * ki + ((row == col) ? 1.f : 0.f);
    Pb[IOFF + idx] = ar * ki + ai * kr;
}

// planes -> interleaved float output (paired 8B stores)
__global__ __launch_bounds__(256)
void finalize_kernel(float* __restrict__ out, const float* __restrict__ src,
                     long long batchStride)
{
    const int b = blockIdx.z;
    const int idx = blockIdx.x * 256 + threadIdx.x;
    const float* s = src + (long long)b * batchStride;
    float* o = out + (long long)b * (2LL * 65536);
    float2 w; w.x = s[idx]; w.y = s[IOFF + idx];
    *(float2*)&o[2 * idx] = w;
}

// ---------------------------------------------------------------------------
// Complex f32 WMMA GEMM: C = alpha*(A.B) + gamma*I per batch.
//   alpha = alphaR * (alphaB ? alphaB[b] : 1), optional conj(A).
// LDS layout: pair-interleaved sX[(k>>1)*PP + 2*col + (k&1)], so each WMMA
// operand (k,k+1 pair) is a single aligned ds_load_b64; PP%64==32 makes the
// two lane-halves hit disjoint bank halves (conflict-free reads).
// Global loads are 16B (float4) per thread per plane, register double-buffered
// across K chunks to overlap with WMMA compute.
// ---------------------------------------------------------------------------
__global__ __launch_bounds__(256)
void cgemm_wmma(const float* __restrict__ Abase, long long strideA,
                const float* __restrict__ Bbase, long long strideB,
                float* __restrict__ Cbase, long long strideC,
                const double* __restrict__ alphaB,
                float alphaR, float gamma, int conjA)
{
    __shared__ float sAr[8 * PP], sAi[8 * PP];
    __shared__ float sBr[8 * PP], sBi[8 * PP];

    const int b = blockIdx.z;
    const float* A = Abase + (long long)b * strideA;
    const float* B = Bbase + (long long)b * strideB;
    float* C = Cbase + (long long)b * strideC;
    const int m0 = blockIdx.y * TM;
    const int n0 = blockIdx.x * TN;

    float ax = alphaR, ay = 0.f;
    if (alphaB) { ax = alphaR * (float)alphaB[2 * b]; ay = alphaR * (float)alphaB[2 * b + 1]; }
    const float csign = conjA ? -1.f : 1.f;

    const int tid  = threadIdx.x;
    const int wave = tid >> 5;
    const int lane = tid & 31;
    const int wm = (wave >> 2) * 32;       // 0 | 32
    const int wn = (wave & 3) * 16;        // 0,16,32,48
    const int hsel = (lane < 16) ? 0 : 1;  // pair-row select for lane half
    const int mrow = lane & 15;
    const int ncol = lane & 15;

    // loader assignment
    const int mA = tid >> 2;               // 0..63
    const int kA = (tid & 3) << 2;         // 0,4,8,12
    const int kB = tid >> 4;               // 0..15
    const int nB = (tid & 15) << 2;        // 0,4,...,60

    const float* Aptr = A + (m0 + mA) * UDIM + kA;
    const float* Bptr = B + kB * UDIM + n0 + nB;
    const int baseA = (kA >> 1) * PP + 2 * mA;         // pairs kA/2, kA/2+1
    const int baseB = (kB >> 1) * PP + (kB & 1);

    v8f accR0 = {}, accS0 = {}, accI0 = {};
    v8f accR1 = {}, accS1 = {}, accI1 = {};

    // preload chunk 0
    float4 arv = *(const float4*)(Aptr);
    float4 aiv = *(const float4*)(Aptr + IOFF);
    float4 brv = *(const float4*)(Bptr);
    float4 biv = *(const float4*)(Bptr + IOFF);

    for (int k0 = 0; k0 < UDIM; k0 += TK) {
        __syncthreads();
        // publish staged registers to LDS (alpha/conj folded into A)
        {
            const float i0 = csign * aiv.x, i1 = csign * aiv.y;
            const float i2 = csign * aiv.z, i3 = csign * aiv.w;
            v2f p0, p1, q0, q1;
            p0.x = arv.x * ax - i0 * ay;  p0.y = arv.y * ax - i1 * ay;
            p1.x = arv.z * ax - i2 * ay;  p1.y = arv.w * ax - i3 * ay;
            q0.x = arv.x * ay + i0 * ax;  q0.y = arv.y * ay + i1 * ax;
            q1.x = arv.z * ay + i2 * ax;  q1.y = arv.w * ay + i3 * ax;
            *(v2f*)&sAr[baseA]      = p0;
            *(v2f*)&sAr[baseA + PP] = p1;
            *(v2f*)&sAi[baseA]      = q0;
            *(v2f*)&sAi[baseA + PP] = q1;
            sBr[baseB + 2 * (nB + 0)] = brv.x;
            sBr[baseB + 2 * (nB + 1)] = brv.y;
            sBr[baseB + 2 * (nB + 2)] = brv.z;
            sBr[baseB + 2 * (nB + 3)] = brv.w;
            sBi[baseB + 2 * (nB + 0)] = biv.x;
            sBi[baseB + 2 * (nB + 1)] = biv.y;
            sBi[baseB + 2 * (nB + 2)] = biv.z;
            sBi[baseB + 2 * (nB + 3)] = biv.w;
        }
        __syncthreads();

        // prefetch next chunk while computing this one
        if (k0 + TK < UDIM) {
            arv = *(const float4*)(Aptr + k0 + TK);
            aiv = *(const float4*)(Aptr + IOFF + k0 + TK);
            brv = *(const float4*)(Bptr + (long long)(k0 + TK) * UDIM);
            biv = *(const float4*)(Bptr + IOFF + (long long)(k0 + TK) * UDIM);
        }

#pragma unroll
        for (int kk = 0; kk < TK; kk += 4) {
            const int p = (kk >> 1) + hsel;
            const int aoff = p * PP + 2 * (wm + mrow);
            const int boff = p * PP + 2 * (wn + ncol);
            v2f ar0 = *(const v2f*)&sAr[aoff];
            v2f ai0 = *(const v2f*)&sAi[aoff];
            v2f ar1 = *(const v2f*)&sAr[aoff + 32];
            v2f ai1 = *(const v2f*)&sAi[aoff + 32];
            v2f br  = *(const v2f*)&sBr[boff];
            v2f bi  = *(const v2f*)&sBi[boff];

            accR0 = __builtin_amdgcn_wmma_f32_16x16x4_f32(false, ar0, false, br, (short)0, accR0, false, false);
            accS0 = __builtin_amdgcn_wmma_f32_16x16x4_f32(false, ai0, false, bi, (short)0, accS0, false, false);
            accI0 = __builtin_amdgcn_wmma_f32_16x16x4_f32(false, ar0, false, bi, (short)0, accI0, false, false);
            accI0 = __builtin_amdgcn_wmma_f32_16x16x4_f32(false, ai0, false, br, (short)0, accI0, false, false);
            accR1 = __builtin_amdgcn_wmma_f32_16x16x4_f32(false, ar1, false, br, (short)0, accR1, false, false);
            accS1 = __builtin_amdgcn_wmma_f32_16x16x4_f32(false, ai1, false, bi, (short)0, accS1, false, false);
            accI1 = __builtin_amdgcn_wmma_f32_16x16x4_f32(false, ar1, false, bi, (short)0, accI1, false, false);
            accI1 = __builtin_amdgcn_wmma_f32_16x16x4_f32(false, ai1, false, br, (short)0, accI1, false, false);
        }
    }

    // epilogue: real = R - S (+ gamma on diagonal), imag = I
    const int n = n0 + wn + ncol;
#pragma unroll
    for (int v = 0; v < 8; ++v) {
        const int mloc = v + ((lane < 16) ? 0 : 8);
        const int m  = m0 + wm + mloc;
        const int m1 = m + 16;
        C[m * UDIM + n]         = accR0[v] - accS0[v] + ((m == n) ? gamma : 0.f);
        C[IOFF + m * UDIM + n]  = accI0[v];
        C[m1 * UDIM + n]        = accR1[v] - accS1[v] + ((m1 == n) ? gamma : 0.f);
        C[IOFF + m1 * UDIM + n] = accI1[v];
    }
}

// ---------------------------------------------------------------------------
// Host orchestration
// ---------------------------------------------------------------------------
static inline void cgemm(hipStream_t st, const float* A, long long sA,
                         const float* B, long long sB, float* C, long long sC,
                         int batches, const double* alphaB, float alphaR,
                         float gamma, int conjA)
{
    dim3 grid(UDIM / TN, UDIM / TM, batches);
    cgemm_wmma<<<grid, 256, 0, st>>>(A, sA, B, sB, C, sC, alphaB, alphaR, gamma, conjA);
}

static inline void axpby(hipStream_t st, float* D, const float* X, const float* Y,
                         float ar, float ai, float br, float bi, float g)
{
    caxpby_kernel<<<256, 256, 0, st>>>(D, X, Y, ar, ai, br, bi, g);
}

extern "C" void kernel_launch(void* const* d_in, const int* in_sizes, int n_in,
                              void* d_out, int out_size, void* d_ws, size_t ws_size,
                              hipStream_t stream)
{
    (void)in_sizes; (void)n_in; (void)out_size; (void)ws_size;
    const double* x     = (const double*)d_in[0];
    const double* A_r   = (const double*)d_in[1];
    const double* A_i   = (const double*)d_in[2];
    const double* cA_r  = (const double*)d_in[3];
    const double* cA_i  = (const double*)d_in[4];
    const double* Ew_r  = (const double*)d_in[5];
    const double* Ew_i  = (const double*)d_in[6];
    const double* uw_r  = (const double*)d_in[7];
    const double* uw_i  = (const double*)d_in[8];

    float* out = (float*)d_out;
    float* out_update = out;                                  // 256
    float* out_expt   = out + 256;                            // 2*256^3
    float* out_newuw  = out_expt + 2LL * 256 * 256 * 256;     // 2*65536
    float* out_newEw  = out_newuw + 2 * 65536;                // 2*65536

    // workspace layout
    double* wsd   = (double*)d_ws;
    double* unitW = wsd;          // 512
    double* zscal = wsd + 512;    // 512
    double* logdf = wsd + 1024;   // 512
    double* scal  = wsd + 1536;   // 8
    float* wsf = (float*)((char*)d_ws + 32768);
    float* mEw  = wsf + 0LL  * MATF;
    float* mY   = wsf + 1LL  * MATF;
    float* mZ   = wsf + 2LL  * MATF;
    float* mT   = wsf + 3LL  * MATF;
    float* mW   = wsf + 4LL  * MATF;
    float* mY2  = wsf + 5LL  * MATF;
    float* mZ2  = wsf + 6LL  * MATF;
    float* mG   = wsf + 7LL  * MATF;
    float* mG2  = wsf + 8LL  * MATF;
    float* mL   = wsf + 9LL  * MATF;
    float* mK   = wsf + 10LL * MATF;
    float* mXuw = wsf + 11LL * MATF;
    float* mE1  = wsf + 12LL * MATF;
    float* mE2  = wsf + 13LL * MATF;
    float* Pbuf = wsf + 16LL * MATF;                 // 256 * MATF
    float* Qbuf = Pbuf + 256LL * MATF;               // 256 * MATF

    // ---- prologue (f64) ----
    prep_kernel<<<1, 256, 0, stream>>>(x, A_r, A_i, cA_r, cA_i,
                                       unitW, zscal, logdf, out_update);
    build_uw_kernel<<<256, 256, 0, stream>>>(uw_r, uw_i, logdf, out_newuw, mXuw);
    convert_kernel<<<256, 256, 0, stream>>>(Ew_r, Ew_i, mEw);
    trace_kernel<<<1, 256, 0, stream>>>(Ew_r, Ew_i, scal);

    // ---- logm(Ew): normalize, chained NS square roots, Gregory-Horner ----
    scale_dev_kernel<<<512, 256, 0, stream>>>(mY, mEw, scal + 1);  // Y = Ew / c
    for (int lvl = 0; lvl < SLOG; ++lvl) {
        axpby(stream, mZ, nullptr, nullptr, 0, 0, 0, 0, 1.f);      // Z = I
        for (int it = 0; it < NSIT; ++it) {
            cgemm(stream, mZ, 0, mY, 0, mT, 0, 1, nullptr, 1.f, 0.f, 0);   // T = Z.Y
            axpby(stream, mW, mT, nullptr, -0.5f, 0, 0, 0, 1.5f);          // W = 1.5I - 0.5T
            cgemm(stream, mY, 0, mW, 0, mY2, 0, 1, nullptr, 1.f, 0.f, 0);  // Y = Y.W
            cgemm(stream, mW, 0, mZ, 0, mZ2, 0, 1, nullptr, 1.f, 0.f, 0);  // Z = W.Z
            float* t1 = mY; mY = mY2; mY2 = t1;
            float* t2 = mZ; mZ = mZ2; mZ2 = t2;
        }
    }
    axpby(stream, mT, mY, nullptr, 1.f, 0, 0, 0, -1.f);            // X = Y - I  (in mT)
    axpby(stream, mG, mT, nullptr, -0.25f, 0, 0, 0, 1.f / 3.f);    // G = -X/4 + I/3
    cgemm(stream, mT, 0, mG, 0, mG2, 0, 1, nullptr, 1.f, -0.5f, 0);// G = X.G - I/2
    { float* t = mG; mG = mG2; mG2 = t; }
    cgemm(stream, mT, 0, mG, 0, mG2, 0, 1, nullptr, 1.f, 1.f, 0);  // G = X.G + I
    { float* t = mG; mG = mG2; mG2 = t; }
    cgemm(stream, mT, 0, mG, 0, mL, 0, 1, nullptr, (float)(1 << SLOG), 0.f, 0); // L = 2^s X.G
    add_logc_kernel<<<1, 256, 0, stream>>>(mL, scal);              // L += log(c) I
    build_K_kernel<<<256, 256, 0, stream>>>(mL, mK);               // K = conj(L)-conj(L)^T

    // ---- batched expm(z_b * K): Horner Taylor + S2 squarings ----
    {
        dim3 g(256, 1, 256);
        init_P_kernel<<<g, 256, 0, stream>>>(Pbuf, mK, zscal, 1.f / (float)MTAY);
    }
    float* P = Pbuf; float* Q = Qbuf;
    for (int k = MTAY - 1; k >= 1; --k) {
        // Q[b] = (z_b/k) K . P[b] + I   (A shared, per-batch complex alpha)
        cgemm(stream, mK, 0, P, MATF, Q, MATF, 256, zscal, 1.f / (float)k, 1.f, 0);
        float* t = P; P = Q; Q = t;
    }
    for (int s = 0; s < S2_SQ; ++s) {
        cgemm(stream, P, MATF, P, MATF, Q, MATF, 256, nullptr, 1.f, 0.f, 0);
        float* t = P; P = Q; Q = t;
    }
    {
        dim3 g(256, 1, 256);
        finalize_kernel<<<g, 256, 0, stream>>>(out_expt, P, (long long)MATF);
    }

    // ---- expm(new_uw) and new_Ew = conj(Ew) . expm(new_uw) ----
    axpby(stream, mE1, mXuw, nullptr, 1.f / (float)MTAY, 0, 0, 0, 1.f);
    for (int k = MTAY - 1; k >= 1; --k) {
        cgemm(stream, mXuw, 0, mE1, 0, mE2, 0, 1, nullptr, 1.f / (float)k, 1.f, 0);
        float* t = mE1; mE1 = mE2; mE2 = t;
    }
    for (int s = 0; s < S3_SQ; ++s) {
        cgemm(stream, mE1, 0, mE1, 0, mE2, 0, 1, nullptr, 1.f, 0.f, 0);
        float* t = mE1; mE1 = mE2; mE2 = t;
    }
    cgemm(stream, mEw, 0, mE1, 0, mT, 0, 1, nullptr, 1.f, 0.f, 1 /*conjA*/);
    {
        dim3 g(256, 1, 1);
        finalize_kernel<<<g, 256, 0, stream>>>(out_newEw, mT, 0LL);
    }
}